// SpatioTemporalFusion_36601711296535
// MI455X (gfx1250) — compile-verified
//
#include <hip/hip_runtime.h>
#include <math.h>

typedef float v2f __attribute__((ext_vector_type(2)));
typedef float v8f __attribute__((ext_vector_type(8)));

#define HH 64
#define WW 64
#define HWP (HH * WW)   // 4096 pixels
#define CC  8           // channels per K-chunk (8*9 = 72 K-rows staged in LDS)
#define NT  32          // pixel tile per block (two 16-px WMMA subtiles)

// ---------------------------------------------------------------------------
// Build feat = concat(frameA, frameB) along channels.
// inputs: (4,128,HW).  feat: (2,256,HW).  selA/selB pick even(0)/odd(1) frames.
// ---------------------------------------------------------------------------
__global__ void concat_feat(const float* __restrict__ in, float* __restrict__ feat,
                            int selA, int selB) {
  int idx = blockIdx.x * 256 + threadIdx.x;
  const int total = 2 * 256 * HWP;
  if (idx >= total) return;
  int p = idx % HWP;
  int c = (idx / HWP) % 256;
  int b = idx / (HWP * 256);
  int sel = (c < 128) ? selA : selB;
  int cc = c & 127;
  feat[idx] = in[((size_t)(b * 2 + sel) * 128 + cc) * HWP + p];
}

// ---------------------------------------------------------------------------
// Fused (deformable) im2col-sample + GEMM on V_WMMA_F32_16X16X4_F32.
//   out[b,o,p] = sum_{c,k9} wgt[o, c*9+k9] * bilinear(x[b,c], p, k9, offset)
// offset == nullptr -> plain 3x3 conv (zero pad).
// Block = 128 threads = 4 waves. Tile: 64 output channels (4 x 16-row M
// tiles, one per wave) x 32 pixels (two 16-px N subtiles per wave, sharing
// one A fragment -> 2 WMMAs per weight load).
// The sampled B matrix is staged chunk-by-chunk in LDS, PAIR-INTERLEAVED:
// sB[pairRow][px] : v2f holding K-rows (2r, 2r+1) for one pixel, so a WMMA
// B fragment is a single aligned ds_load_b64 straight into the operand pair.
// Bilinear tables depend only on (tap, pixel), built once per block.
// ---------------------------------------------------------------------------
template <int CIN, int COUT, int RELU>
__global__ __launch_bounds__(128)
void deform_gemm(const float* __restrict__ x,
                 const float* __restrict__ offset,   // (B, 18, HW) or nullptr
                 const float* __restrict__ wgt,      // (COUT, CIN*9)
                 float* __restrict__ out,            // (B, COUT, HW)
                 int bmul, int bsel) {
  __shared__ int4   s_idx[9 * NT];          // taps x px: clamped gather indices
  __shared__ float4 s_wt[9 * NT];           // masked bilinear weights
  __shared__ v2f    sB[(CC * 9 / 2) * NT];  // 36 pair-rows x 32 px

  const int b   = blockIdx.z;
  const int n0  = blockIdx.x * NT;          // pixel tile base
  const int tid = threadIdx.x;

  // ---- bilinear table (9 taps x 32 pixels) ----
  for (int e = tid; e < 9 * NT; e += 128) {
    int k9  = e >> 5;
    int pix = e & (NT - 1);
    int p   = n0 + pix;
    int h   = p / WW, w = p % WW;
    int ki  = k9 / 3, kj = k9 % 3;
    float dy = 0.f, dx = 0.f;
    if (offset) {
      dy = offset[((size_t)b * 18 + 2 * k9 + 0) * HWP + p];
      dx = offset[((size_t)b * 18 + 2 * k9 + 1) * HWP + p];
    }
    float py = (float)(h + ki - 1) + dy;
    float px = (float)(w + kj - 1) + dx;
    float y0f = floorf(py), x0f = floorf(px);
    int   y0  = (int)y0f,  x0  = (int)x0f;
    float fy  = py - y0f,  fx  = px - x0f;
    float wy0 = 1.f - fy,  wx0 = 1.f - fx;
    float w4[4] = { wy0 * wx0, wy0 * fx, fy * wx0, fy * fx };
    int   yy[4] = { y0, y0, y0 + 1, y0 + 1 };
    int   xx[4] = { x0, x0 + 1, x0, x0 + 1 };
    int   gi[4];
    float gw[4];
#pragma unroll
    for (int q = 0; q < 4; ++q) {
      bool valid = (yy[q] >= 0) && (yy[q] < HH) && (xx[q] >= 0) && (xx[q] < WW);
      int yc = yy[q] < 0 ? 0 : (yy[q] > HH - 1 ? HH - 1 : yy[q]);
      int xc = xx[q] < 0 ? 0 : (xx[q] > WW - 1 ? WW - 1 : xx[q]);
      gi[q] = yc * WW + xc;
      gw[q] = valid ? w4[q] : 0.f;
    }
    s_idx[e] = make_int4(gi[0], gi[1], gi[2], gi[3]);
    s_wt[e]  = make_float4(gw[0], gw[1], gw[2], gw[3]);
  }

  const int wave = tid >> 5;
  const int lane = tid & 31;
  const int half = lane >> 4;               // K-pair select for A/B fragments
  const int l16  = lane & 15;
  const int m0   = (blockIdx.y * 4 + wave) * 16;
  const bool mvalid = (m0 < COUT);          // wave-uniform

  const int Ktot = CIN * 9;
  const float* xbase = x + (size_t)(b * bmul + bsel) * CIN * HWP;

  const int  mA  = m0 + l16;                // A-fragment row for this lane
  const bool mAv = (COUT % 16 == 0) ? true : (mA < COUT);
  const float2* wrow2 =
      (const float2*)(wgt + (size_t)(mAv ? mA : 0) * Ktot);

  v8f acc0 = {0.f, 0.f, 0.f, 0.f, 0.f, 0.f, 0.f, 0.f};
  v8f acc1 = {0.f, 0.f, 0.f, 0.f, 0.f, 0.f, 0.f, 0.f};

#pragma unroll 1
  for (int c0 = 0; c0 < CIN; c0 += CC) {
    __syncthreads();   // prev chunk consumed (also covers table build on iter 0)

    // ---- cooperative fill: 36 pair-rows x 32 px, 9 pairs (18 samples)/thread
#pragma unroll
    for (int it = 0; it < (CC * 9 * NT) / 256; ++it) {
      int q  = tid + it * 128;         // pair-slot index
      int pr = q >> 5;                 // pair row 0..35
      int px = q & (NT - 1);
      v2f val;
#pragma unroll
      for (int j = 0; j < 2; ++j) {
        int row = pr * 2 + j;          // K-row  (= c_local*9 + k9)
        int cl  = row / 9;
        int k9  = row - cl * 9;
        int e   = (k9 << 5) + px;
        const float* plane = xbase + (size_t)(c0 + cl) * HWP;
        int4   gi = s_idx[e];
        float4 gw = s_wt[e];
        float v = gw.x * plane[gi.x] + gw.y * plane[gi.y]
                + gw.z * plane[gi.z] + gw.w * plane[gi.w];
        if (j == 0) val.x = v; else val.y = v;
      }
      sB[q] = val;                     // one ds_store_b64
    }
    __syncthreads();

    // ---- WMMA over this chunk: 72 K-rows / 4 = 18 steps, 2 WMMAs each ----
    if (mvalid) {
      const int kk = c0 * 9;
#pragma unroll
      for (int st = 0; st < (CC * 9) / 4; ++st) {
        const int kl = st * 4 + half * 2;        // local K row (even)
        float2 t = wrow2[(kk + kl) >> 1];        // one b64 weight load
        v2f a;
        if constexpr (COUT % 16 == 0) {
          a.x = t.x; a.y = t.y;                  // full tile: no mask needed
        } else {
          a.x = mAv ? t.x : 0.f;
          a.y = mAv ? t.y : 0.f;
        }
        const int pbase = (st * 2 + half) * NT;  // v2f slots per pair-row = 32
        v2f bf0 = sB[pbase + l16];               // ds_load_b64, WMMA-ready
        v2f bf1 = sB[pbase + 16 + l16];
        acc0 = __builtin_amdgcn_wmma_f32_16x16x4_f32(false, a, false, bf0,
                                                     (short)0, acc0, false, false);
        acc1 = __builtin_amdgcn_wmma_f32_16x16x4_f32(false, a, false, bf1,
                                                     (short)0, acc1, false, false);
      }
    }
  }

  // C/D layout: VGPR v -> M = m0 + v + half*8, N = n0 + l16 (+16 for acc1)
  if (mvalid) {
#pragma unroll
    for (int v = 0; v < 8; ++v) {
      int mg = m0 + v + half * 8;
      if (COUT % 16 == 0 || mg < COUT) {
        float r0 = acc0[v];
        float r1 = acc1[v];
        if (RELU) { r0 = fmaxf(r0, 0.f); r1 = fmaxf(r1, 0.f); }
        size_t base = ((size_t)b * COUT + mg) * HWP + n0 + l16;
        out[base]      = r0;
        out[base + 16] = r1;
      }
    }
  }
}

// ---------------------------------------------------------------------------
// Per-pixel cosine similarity -> softmax over {Wx, Wy} -> blend deform0/1.
// ---------------------------------------------------------------------------
__global__ void blend_kernel(const float* __restrict__ d0, const float* __restrict__ d1,
                             const float* __restrict__ w0, const float* __restrict__ w1,
                             float* __restrict__ out) {
  int idx = blockIdx.x * 256 + threadIdx.x;
  const int total = 2 * 128 * HWP;
  if (idx >= total) return;
  int p = idx % HWP;
  int b = idx / (HWP * 128);
  float a  = w0[b * HWP + p];
  float bb = w1[b * HWP + p];
  const float EPS = 1e-8f;
  float wx = (a * bb)  / sqrtf(fmaxf((a * a) * (bb * bb),   EPS * EPS));
  float wy = (bb * bb) / sqrtf(fmaxf((bb * bb) * (bb * bb), EPS * EPS));
  float m  = fmaxf(wx, wy);
  float e0 = __expf(wx - m), e1 = __expf(wy - m);
  float inv = 1.f / (e0 + e1);
  out[idx] = d0[idx] * (e0 * inv) + d1[idx] * (e1 * inv);
}

// ---------------------------------------------------------------------------
template <int CIN, int COUT, int RELU>
static inline void launch_dg(const float* x, const float* off, const float* w,
                             float* out, int bmul, int bsel, hipStream_t s) {
  constexpr int mtiles = (COUT + 15) / 16;
  dim3 grid(HWP / NT, (mtiles + 3) / 4, 2);
  deform_gemm<CIN, COUT, RELU><<<grid, 128, 0, s>>>(x, off, w, out, bmul, bsel);
}

extern "C" void kernel_launch(void* const* d_in, const int* in_sizes, int n_in,
                              void* d_out, int out_size, void* d_ws, size_t ws_size,
                              hipStream_t stream) {
  (void)in_sizes; (void)n_in; (void)out_size; (void)ws_size;

  const float* inputs   = (const float*)d_in[2];   // (4,128,64,64)
  const float* off_w[4] = { (const float*)d_in[3], (const float*)d_in[4],
                            (const float*)d_in[5], (const float*)d_in[6] };
  const float* def_w[3] = { (const float*)d_in[7], (const float*)d_in[8],
                            (const float*)d_in[9] };
  const float* pred_w   = (const float*)d_in[10];
  const float* s_w[3]   = { (const float*)d_in[11], (const float*)d_in[12],
                            (const float*)d_in[13] };

  float* ws      = (float*)d_ws;
  float* featA   = ws;                                 // 2*256*HWP
  float* featB   = featA   + (size_t)2 * 256 * HWP;
  float* offb    = featB   + (size_t)2 * 256 * HWP;    // 2*18*HWP
  float* deform0 = offb    + (size_t)2 * 18  * HWP;    // 2*128*HWP
  float* deform1 = deform0 + (size_t)2 * 128 * HWP;
  float* h1      = deform1 + (size_t)2 * 128 * HWP;    // 2*64*HWP
  float* h2      = h1      + (size_t)2 * 64  * HWP;    // 2*32*HWP
  float* wb0     = h2      + (size_t)2 * 32  * HWP;    // 2*HWP
  float* wb1     = wb0     + (size_t)2 * HWP;

  const int cat_total = 2 * 256 * HWP;

  for (int branch = 0; branch < 2; ++branch) {
    int selA = (branch == 0) ? 0 : 1;   // branch0: (x,y)   branch1: (y,y)

    concat_feat<<<(cat_total + 255) / 256, 256, 0, stream>>>(inputs, featA, selA, 1);

    float* fa = featA;
    float* fb = featB;
    for (int i = 0; i < 3; ++i) {
      launch_dg<256, 18, 0>(fa, nullptr, off_w[i], offb, 1, 0, stream);
      launch_dg<256, 256, 0>(fa, offb, def_w[i], fb, 1, 0, stream);
      float* t = fa; fa = fb; fb = t;
    }
    launch_dg<256, 18, 0>(fa, nullptr, off_w[3], offb, 1, 0, stream);

    float* dfm = (branch == 0) ? deform0 : deform1;
    launch_dg<128, 128, 0>(inputs, offb, pred_w, dfm, 2, selA, stream);

    launch_dg<128, 64, 1>(dfm, nullptr, s_w[0], h1, 1, 0, stream);
    launch_dg<64, 32, 1>(h1, nullptr, s_w[1], h2, 1, 0, stream);
    launch_dg<32, 1, 1>(h2, nullptr, s_w[2], (branch == 0) ? wb0 : wb1,
                        1, 0, stream);
  }

  const int out_total = 2 * 128 * HWP;
  blend_kernel<<<(out_total + 255) / 256, 256, 0, stream>>>(
      deform0, deform1, wb0, wb1, (float*)d_out);
}